// PointSetKNOHead_7430293422131
// MI455X (gfx1250) — compile-verified
//
#include <hip/hip_runtime.h>
#include <math.h>

typedef __attribute__((ext_vector_type(2))) float v2f;
typedef __attribute__((ext_vector_type(8))) float v8f;
typedef __attribute__((ext_vector_type(4))) unsigned int u32x4;
typedef __attribute__((ext_vector_type(4))) int i32x4;
typedef __attribute__((ext_vector_type(8))) int i32x8;

#define SQ5 2.23606797749979f
#define FIVE_THIRDS 1.6666666666666667f
#define JITTER 1e-6f
#define USE_TDM 1

static constexpr int N1 = 4096;   // x points
static constexpr int NQ = 1024;   // q points
static constexpr int NY = 4096;   // y points
static constexpr int CH = 64;     // channels
static constexpr int DEPTH = 4;
static constexpr int NB = 64;     // Cholesky block

// ---------------- device helpers ----------------

__device__ __forceinline__ float softplusf(float x) {
  return fmaxf(x, 0.f) + log1pf(__expf(-fabsf(x)));
}

__device__ __forceinline__ float geluf(float x) {
  // JAX default gelu (approximate=True, tanh form)
  float x3 = x * x * x;
  return 0.5f * x * (1.f + tanhf(0.7978845608028654f * (x + 0.044715f * x3)));
}

__device__ __forceinline__ float matern_r(float r, float s) {
  float rs = r * s;
  float a = SQ5 * rs;
  return (1.f + a + FIVE_THIRDS * rs * rs) * __expf(-a);
}

__device__ __forceinline__ float dist2d(float ax, float ay, float bx, float by) {
  float dx = ax - bx, dy = ay - by;
  return sqrtf(fmaxf(dx * dx + dy * dy, 0.f));
}

__device__ __forceinline__ v8f wmma4(v2f a, v2f b, v8f c) {
  // D = A(16x4,f32) * B(4x16,f32) + C(16x16,f32)
  return __builtin_amdgcn_wmma_f32_16x16x4_f32(false, a, false, b, (short)0, c,
                                               false, false);
}

#if USE_TDM
// TDM: DMA a 64x64 f32 tile (row stride = stride elements) from global into
// LDS at byte offset lds_off, inserting 4 pad DWORDs every 64 DWORDs
// (matches the [64][68] padded LDS tiles). D# packing per CDNA5 ISA 8.3/8.4.
__device__ __forceinline__ void tdm_load_tile64(unsigned lds_off,
                                                const float* gptr,
                                                unsigned stride_elems) {
  unsigned long long ga = (unsigned long long)(size_t)gptr;
  u32x4 g0;
  g0[0] = 1u;                                   // count=1, user descriptor
  g0[1] = lds_off;                              // lds_addr (bytes)
  g0[2] = (unsigned)ga;                         // global_addr[31:0]
  g0[3] = (unsigned)((ga >> 32) & 0x01FFFFFFull) | (2u << 30);  // type=2
  i32x8 g1;
  // data_size=4B (2), pad_enable, pad_interval=64 DW (5), pad_amount=4 DW (3)
  g1[0] = (int)((2u << 16) | (1u << 20) | (5u << 22) | (3u << 25));
  g1[1] = (int)(64u << 16);                     // tensor_dim0 = 64 (lo16)
  g1[2] = (int)(64u << 16);                     // dim0 hi=0, tensor_dim1=64 lo16
  g1[3] = (int)(64u << 16);                     // dim1 hi=0, tile_dim0=64
  g1[4] = (int)(64u);                           // tile_dim1=64, tile_dim2=0
  g1[5] = (int)stride_elems;                    // tensor_dim0_stride[31:0]
  g1[6] = 0;                                    // stride hi, dim1_stride lo
  g1[7] = 0;
  i32x4 z4 = {0, 0, 0, 0};
  i32x8 z8 = {0, 0, 0, 0, 0, 0, 0, 0};
  __builtin_amdgcn_tensor_load_to_lds(g0, g1, z4, z4, z8, 0);
}
#endif

// ---------------- kernel-matrix builders ----------------

__global__ void lift_kernel(const float* __restrict__ f_x,
                            const float* __restrict__ X,
                            const float* __restrict__ W,
                            const float* __restrict__ b, float* __restrict__ FX) {
  int idx = blockIdx.x * 256 + threadIdx.x;  // N1*CH
  int n = idx >> 6, c = idx & 63;
  FX[idx] = f_x[n] * W[c * 3 + 0] + X[2 * n] * W[c * 3 + 1] +
            X[2 * n + 1] * W[c * 3 + 2] + b[c];
}

__global__ void build_kxx(float* __restrict__ K, const float* __restrict__ X,
                          const float* __restrict__ s_raw, int N) {
  int j = blockIdx.x * 16 + threadIdx.x;
  int i = blockIdx.y * 16 + threadIdx.y;
  float s = softplusf(s_raw[0]);
  float r = dist2d(X[2 * i], X[2 * i + 1], X[2 * j], X[2 * j + 1]);
  float v = matern_r(r, s);
  if (i == j) v += JITTER;
  K[(size_t)i * N + j] = v;
}

__global__ void build_kxq(float* __restrict__ K, const float* __restrict__ X,
                          const float* __restrict__ Q,
                          const float* __restrict__ s_raw) {
  int m = blockIdx.x * 16 + threadIdx.x;  // q index
  int n = blockIdx.y * 16 + threadIdx.y;  // x index
  float s = softplusf(s_raw[0]);
  float r = dist2d(X[2 * n], X[2 * n + 1], Q[2 * m], Q[2 * m + 1]);
  K[(size_t)n * NQ + m] = matern_r(r, s);
}

__global__ void build_rqq(float* __restrict__ R, const float* __restrict__ Q) {
  int j = blockIdx.x * 16 + threadIdx.x;
  int i = blockIdx.y * 16 + threadIdx.y;
  R[(size_t)i * NQ + j] = dist2d(Q[2 * i], Q[2 * i + 1], Q[2 * j], Q[2 * j + 1]);
}

__global__ void build_kqq(float* __restrict__ K, const float* __restrict__ R,
                          const float* __restrict__ s_raw) {
  int j = blockIdx.x * 16 + threadIdx.x;
  int i = blockIdx.y * 16 + threadIdx.y;
  float s = softplusf(s_raw[0]);
  float v = matern_r(R[(size_t)i * NQ + j], s);
  if (i == j) v += JITTER;
  K[(size_t)i * NQ + j] = v;
}

__global__ void build_kyq(float* __restrict__ K, const float* __restrict__ Y,
                          const float* __restrict__ Q,
                          const float* __restrict__ s_raw) {
  int j = blockIdx.x * 16 + threadIdx.x;  // q index
  int m = blockIdx.y * 16 + threadIdx.y;  // y index
  float s = softplusf(s_raw[0]);
  float r = dist2d(Y[2 * m], Y[2 * m + 1], Q[2 * j], Q[2 * j + 1]);
  K[(size_t)m * NQ + j] = matern_r(r, s);
}

// ---------------- blocked Cholesky (lower, in place) ----------------

__global__ void chol_diag(float* __restrict__ A, int N, int k) {
  __shared__ float T[64][65];
  __shared__ float colj[64];
  int t = threadIdx.x;  // 64 threads
  for (int j = 0; j < 64; j++) T[t][j] = A[(size_t)(k + t) * N + k + j];
  for (int j = 0; j < 64; j++) {
    __syncthreads();
    if (t == j) T[j][j] = sqrtf(fmaxf(T[j][j], 1e-20f));
    __syncthreads();
    float l = (t > j) ? (T[t][j] / T[j][j]) : 0.f;
    colj[t] = l;
    __syncthreads();
    if (t > j) {
      T[t][j] = l;
      for (int c = j + 1; c <= t; c++) T[t][c] -= l * colj[c];
    }
  }
  __syncthreads();
  for (int j = 0; j <= t; j++) A[(size_t)(k + t) * N + k + j] = T[t][j];
}

__global__ void chol_panel(float* __restrict__ A, int N, int k) {
  __shared__ float L11[64][65];
  int t = threadIdx.x;  // 64
  for (int j = 0; j < 64; j++) L11[t][j] = A[(size_t)(k + t) * N + k + j];
  __syncthreads();
  int r = k + 64 + blockIdx.x * 64 + t;
  float* Ar = &A[(size_t)r * N + k];
  float a[64];
#pragma unroll
  for (int j = 0; j < 64; j++) a[j] = Ar[j];
#pragma unroll
  for (int j = 0; j < 64; j++) {
    float s = a[j];
    for (int p = 0; p < j; p++) s -= a[p] * L11[j][p];
    a[j] = s / L11[j][j];
  }
#pragma unroll
  for (int j = 0; j < 64; j++) Ar[j] = a[j];
}

// Trailing SYRK update: A22 -= L21 * L21^T (lower macro-tiles), WMMA f32.
// TDM stages both 64x64 panels into padded LDS; each wave runs two
// independent accumulator chains to hide WMMA RAW latency.
template <int LDN>
__global__ void syrk_tile(float* __restrict__ A, int kdiag) {
  int idx = blockIdx.x;
  int bi = (int)((sqrtf(8.f * (float)idx + 1.f) - 1.f) * 0.5f);
  while ((bi + 1) * (bi + 2) / 2 <= idx) bi++;
  while (bi * (bi + 1) / 2 > idx) bi--;
  int bj = idx - bi * (bi + 1) / 2;
  int base = kdiag + NB;
  int gi0 = base + bi * 64, gj0 = base + bj * 64;

  __shared__ float Ablk[64][68];
  __shared__ float Bblk[64][68];
  int t = threadIdx.x;  // 256

#if USE_TDM
  if (t < 32) {  // wave 0 drives the Tensor Data Mover
    tdm_load_tile64((unsigned)(size_t)(void*)&Ablk[0][0],
                    &A[(size_t)gi0 * LDN + kdiag], LDN);
    tdm_load_tile64((unsigned)(size_t)(void*)&Bblk[0][0],
                    &A[(size_t)gj0 * LDN + kdiag], LDN);
    __builtin_amdgcn_s_wait_tensorcnt(0);
  }
#else
  for (int i = 0; i < 4; i++) {
    int fid = t + i * 256;
    int row = fid >> 4, c4 = (fid & 15) << 2;
    float4 va = *(const float4*)&A[(size_t)(gi0 + row) * LDN + kdiag + c4];
    Ablk[row][c4 + 0] = va.x; Ablk[row][c4 + 1] = va.y;
    Ablk[row][c4 + 2] = va.z; Ablk[row][c4 + 3] = va.w;
    float4 vb = *(const float4*)&A[(size_t)(gj0 + row) * LDN + kdiag + c4];
    Bblk[row][c4 + 0] = vb.x; Bblk[row][c4 + 1] = vb.y;
    Bblk[row][c4 + 2] = vb.z; Bblk[row][c4 + 3] = vb.w;
  }
#endif
  __syncthreads();

  int wv = t >> 5, lane = t & 31, hf = lane >> 4, l16 = lane & 15;
  int ti0 = wv >> 2, tj0 = wv & 3;
  int ti1 = (wv + 8) >> 2, tj1 = (wv + 8) & 3;
  const float* Ar0 = &Ablk[ti0 * 16 + l16][0];
  const float* Br0 = &Bblk[tj0 * 16 + l16][0];
  const float* Ar1 = &Ablk[ti1 * 16 + l16][0];
  const float* Br1 = &Bblk[tj1 * 16 + l16][0];
  v8f acc0 = {}, acc1 = {};
#pragma unroll
  for (int kk = 0; kk < 16; kk++) {
    int k0 = kk * 4 + hf * 2;
    acc0 = wmma4(*(const v2f*)&Ar0[k0], *(const v2f*)&Br0[k0], acc0);
    acc1 = wmma4(*(const v2f*)&Ar1[k0], *(const v2f*)&Br1[k0], acc1);
  }

  // store (skip strict-upper subtiles on diagonal macro-blocks; uniform)
  {
    if (!(bi == bj && ti0 < tj0)) {
      float* Cp = &A[(size_t)(gi0 + ti0 * 16 + hf * 8) * LDN + gj0 + tj0 * 16 + l16];
      float old[8];
#pragma unroll
      for (int v = 0; v < 8; v++) old[v] = Cp[(size_t)v * LDN];
#pragma unroll
      for (int v = 0; v < 8; v++) Cp[(size_t)v * LDN] = old[v] - acc0[v];
    }
    if (!(bi == bj && ti1 < tj1)) {
      float* Cp = &A[(size_t)(gi0 + ti1 * 16 + hf * 8) * LDN + gj0 + tj1 * 16 + l16];
      float old[8];
#pragma unroll
      for (int v = 0; v < 8; v++) old[v] = Cp[(size_t)v * LDN];
#pragma unroll
      for (int v = 0; v < 8; v++) Cp[(size_t)v * LDN] = old[v] - acc1[v];
    }
  }
}

// ---------------- blocked triangular solves on RHS ----------------

__global__ void rhs_diag_fwd(const float* __restrict__ L, int N, int kb,
                             float* __restrict__ B, int nc) {
  __shared__ float T[64][65];
  int t = threadIdx.x;  // 64
  for (int j = 0; j < 64; j++) T[t][j] = L[(size_t)(kb + t) * N + kb + j];
  __syncthreads();
  if (t < nc) {
    float y[64];
#pragma unroll
    for (int r = 0; r < 64; r++) {
      float s = B[(size_t)(kb + r) * nc + t];
      for (int p = 0; p < r; p++) s -= T[r][p] * y[p];
      y[r] = s / T[r][r];
    }
#pragma unroll
    for (int r = 0; r < 64; r++) B[(size_t)(kb + r) * nc + t] = y[r];
  }
}

__global__ void rhs_diag_bwd(const float* __restrict__ L, int N, int kb,
                             float* __restrict__ B, int nc) {
  __shared__ float T[64][65];
  int t = threadIdx.x;  // 64
  for (int j = 0; j < 64; j++) T[t][j] = L[(size_t)(kb + t) * N + kb + j];
  __syncthreads();
  if (t < nc) {
    float z[64];
#pragma unroll
    for (int r = 63; r >= 0; r--) {
      float s = B[(size_t)(kb + r) * nc + t];
      for (int p = r + 1; p < 64; p++) s -= T[p][r] * z[p];
      z[r] = s / T[r][r];
    }
#pragma unroll
    for (int r = 0; r < 64; r++) B[(size_t)(kb + r) * nc + t] = z[r];
  }
}

// B[kb+64.., 0:64] -= L[kb+64.., kb:kb+64] @ B[kb:kb+64, 0:64]   (WMMA f32)
__global__ void rhs_fwd_update(const float* __restrict__ L, int N, int kb,
                               float* __restrict__ B) {
  int t = threadIdx.x, wv = t >> 5, lane = t & 31, hf = lane >> 4, l16 = lane & 15;
  int r0 = kb + 64 + blockIdx.x * 16;
  v8f acc0 = {}, acc1 = {};
#pragma unroll
  for (int kk = 0; kk < 16; kk += 2) {
    int ka = kk * 4 + hf * 2;
    int kbb = ka + 4;
    v2f a0 = *(const v2f*)&L[(size_t)(r0 + l16) * N + kb + ka];
    v2f b0 = {B[(size_t)(kb + ka) * 64 + wv * 16 + l16],
              B[(size_t)(kb + ka + 1) * 64 + wv * 16 + l16]};
    acc0 = wmma4(a0, b0, acc0);
    v2f a1 = *(const v2f*)&L[(size_t)(r0 + l16) * N + kb + kbb];
    v2f b1 = {B[(size_t)(kb + kbb) * 64 + wv * 16 + l16],
              B[(size_t)(kb + kbb + 1) * 64 + wv * 16 + l16]};
    acc1 = wmma4(a1, b1, acc1);
  }
  int m0 = hf * 8;
#pragma unroll
  for (int v = 0; v < 8; v++)
    B[(size_t)(r0 + m0 + v) * 64 + wv * 16 + l16] -= acc0[v] + acc1[v];
}

// B[0:kb, 0:64] -= L[kb:kb+64, 0:kb]^T @ B[kb:kb+64, 0:64]   (WMMA f32)
__global__ void rhs_bwd_update(const float* __restrict__ L, int N, int kb,
                               float* __restrict__ B) {
  int t = threadIdx.x, wv = t >> 5, lane = t & 31, hf = lane >> 4, l16 = lane & 15;
  int r0 = blockIdx.x * 16;
  v8f acc0 = {}, acc1 = {};
#pragma unroll
  for (int kk = 0; kk < 16; kk += 2) {
    int ka = kk * 4 + hf * 2;
    int kbb = ka + 4;
    v2f a0 = {L[(size_t)(kb + ka) * N + r0 + l16],
              L[(size_t)(kb + ka + 1) * N + r0 + l16]};
    v2f b0 = {B[(size_t)(kb + ka) * 64 + wv * 16 + l16],
              B[(size_t)(kb + ka + 1) * 64 + wv * 16 + l16]};
    acc0 = wmma4(a0, b0, acc0);
    v2f a1 = {L[(size_t)(kb + kbb) * N + r0 + l16],
              L[(size_t)(kb + kbb + 1) * N + r0 + l16]};
    v2f b1 = {B[(size_t)(kb + kbb) * 64 + wv * 16 + l16],
              B[(size_t)(kb + kbb + 1) * 64 + wv * 16 + l16]};
    acc1 = wmma4(a1, b1, acc1);
  }
  int m0 = hf * 8;
#pragma unroll
  for (int v = 0; v < 8; v++)
    B[(size_t)(r0 + m0 + v) * 64 + wv * 16 + l16] -= acc0[v] + acc1[v];
}

// single-RHS updates (stage-3 solve)
__global__ void vec_fwd_update(const float* __restrict__ L, int N, int kb,
                               float* __restrict__ bvec) {
  __shared__ float y[64];
  int t = threadIdx.x;
  if (t < 64) y[t] = bvec[kb + t];
  __syncthreads();
  int i = kb + 64 + blockIdx.x * 256 + t;
  if (i < N) {
    float s = 0.f;
#pragma unroll
    for (int p = 0; p < 64; p++) s += L[(size_t)i * N + kb + p] * y[p];
    bvec[i] -= s;
  }
}

__global__ void vec_bwd_update(const float* __restrict__ L, int N, int kb,
                               float* __restrict__ bvec) {
  __shared__ float z[64];
  int t = threadIdx.x;
  if (t < 64) z[t] = bvec[kb + t];
  __syncthreads();
  int i = blockIdx.x * 256 + t;
  if (i < kb) {
    float s = 0.f;
#pragma unroll
    for (int p = 0; p < 64; p++) s += L[(size_t)(kb + p) * N + i] * z[p];
    bvec[i] -= s;
  }
}

// ---------------- C = A^T * B  (Kxq^T @ U), WMMA f32 ----------------
__global__ void gemm_AtB(const float* __restrict__ Am /*K x NQ*/,
                         const float* __restrict__ Bm /*K x 64*/,
                         float* __restrict__ C /*NQ x 64*/, int K) {
  int t = threadIdx.x, wv = t >> 5, lane = t & 31, hf = lane >> 4, l16 = lane & 15;
  int m0t = blockIdx.x * 16;
  v8f acc0 = {}, acc1 = {};
  for (int kk = 0; kk < K / 4; kk += 2) {
    int ka = kk * 4 + hf * 2;
    int kbb = ka + 4;
    if ((kk & 31) == 0) {
      __builtin_prefetch(&Am[(size_t)(ka + 128) * NQ + m0t + l16], 0, 1);
      __builtin_prefetch(&Bm[(size_t)(ka + 128) * 64 + wv * 16 + l16], 0, 1);
    }
    v2f a0 = {Am[(size_t)ka * NQ + m0t + l16],
              Am[(size_t)(ka + 1) * NQ + m0t + l16]};
    v2f b0 = {Bm[(size_t)ka * 64 + wv * 16 + l16],
              Bm[(size_t)(ka + 1) * 64 + wv * 16 + l16]};
    acc0 = wmma4(a0, b0, acc0);
    v2f a1 = {Am[(size_t)kbb * NQ + m0t + l16],
              Am[(size_t)(kbb + 1) * NQ + m0t + l16]};
    v2f b1 = {Bm[(size_t)kbb * 64 + wv * 16 + l16],
              Bm[(size_t)(kbb + 1) * 64 + wv * 16 + l16]};
    acc1 = wmma4(a1, b1, acc1);
  }
  int m0 = hf * 8;
#pragma unroll
  for (int v = 0; v < 8; v++)
    C[(size_t)(m0t + m0 + v) * 64 + wv * 16 + l16] = acc0[v] + acc1[v];
}

__global__ void gelu_kernel(float* __restrict__ x, int n) {
  int i = blockIdx.x * 256 + threadIdx.x;
  if (i < n) x[i] = geluf(x[i]);
}

// ---------------- fused integral layer ----------------
// out[q,c] = maybe_gelu( conv_b[c] + sum_p fq[q,p] W[c,p]
//                        + sum_m matern(r_qq[q,m], s_c) * fq[m,c] * w[m] )
__global__ void kno_layer(const float* __restrict__ fq_in,
                          float* __restrict__ fq_out,
                          const float* __restrict__ Rqq,
                          const float* __restrict__ Wl,
                          const float* __restrict__ bl,
                          const float* __restrict__ scl_raw,
                          const float* __restrict__ w, int apply_gelu) {
  __shared__ float rrow[NQ];
  __shared__ float sc[64];
  __shared__ float fqrow[64];
  __shared__ float part[4][64];
  int q = blockIdx.x;
  int t = threadIdx.x;  // 256
  {
    const float4* src = (const float4*)&Rqq[(size_t)q * NQ];
    float4 v4 = src[t];
    rrow[t * 4 + 0] = v4.x; rrow[t * 4 + 1] = v4.y;
    rrow[t * 4 + 2] = v4.z; rrow[t * 4 + 3] = v4.w;
  }
  if (t < 64) {
    sc[t] = softplusf(scl_raw[t]);
    fqrow[t] = fq_in[(size_t)q * 64 + t];
  }
  __syncthreads();

  int c = t & 63, mr = t >> 6;
  float acc = 0.f;
  float s = sc[c];
  for (int m = mr; m < NQ; m += 4) {
    float kv = matern_r(rrow[m], s);
    acc += kv * fq_in[(size_t)m * 64 + c] * w[m];
  }
  part[mr][c] = acc;
  __syncthreads();
  if (t < 64) {
    float integ = part[0][t] + part[1][t] + part[2][t] + part[3][t];
    float skip = bl[t];
#pragma unroll
    for (int p = 0; p < 64; p++) skip += fqrow[p] * Wl[t * 64 + p];
    float o = skip + integ;
    fq_out[(size_t)q * 64 + t] = apply_gelu ? geluf(o) : o;
  }
}

// ---------------- dense heads ----------------
__global__ void dense64(const float* __restrict__ in, const float* __restrict__ W,
                        const float* __restrict__ b, float* __restrict__ out,
                        int act) {
  int idx = blockIdx.x * 256 + threadIdx.x;
  int q = idx >> 6, c = idx & 63;
  float s = b[c];
#pragma unroll
  for (int p = 0; p < 64; p++) s += in[(size_t)q * 64 + p] * W[c * 64 + p];
  out[idx] = act ? geluf(s) : s;
}

__global__ void proj1(const float* __restrict__ in, const float* __restrict__ W,
                      const float* __restrict__ b, float* __restrict__ h) {
  int q = blockIdx.x * 256 + threadIdx.x;
  float s = b[0];
#pragma unroll
  for (int p = 0; p < 64; p++) s += in[(size_t)q * 64 + p] * W[p];
  h[q] = s;
}

__global__ void final_out(const float* __restrict__ Kyq,
                          const float* __restrict__ v, float* __restrict__ out) {
  __shared__ float red[256];
  int m = blockIdx.x, t = threadIdx.x;
  float s = 0.f;
  for (int j = t; j < NQ; j += 256) s += Kyq[(size_t)m * NQ + j] * v[j];
  red[t] = s;
  __syncthreads();
  for (int o = 128; o > 0; o >>= 1) {
    if (t < o) red[t] += red[t + o];
    __syncthreads();
  }
  if (t == 0) out[m] = red[0];
}

// ---------------- host orchestration ----------------

template <int LDN>
static void cholesky_t(float* A, hipStream_t stream) {
  for (int k = 0; k < LDN; k += NB) {
    chol_diag<<<1, 64, 0, stream>>>(A, LDN, k);
    int M = LDN - k - NB;
    if (M > 0) {
      chol_panel<<<M / 64, 64, 0, stream>>>(A, LDN, k);
      int T = M / 64;
      syrk_tile<LDN><<<T * (T + 1) / 2, 256, 0, stream>>>(A, k);
    }
  }
}

static void chol_solve64(float* A, int N, float* B, hipStream_t stream) {
  for (int kb = 0; kb < N; kb += NB) {
    rhs_diag_fwd<<<1, 64, 0, stream>>>(A, N, kb, B, 64);
    int M = N - kb - NB;
    if (M > 0) rhs_fwd_update<<<M / 16, 128, 0, stream>>>(A, N, kb, B);
  }
  for (int kb = N - NB; kb >= 0; kb -= NB) {
    rhs_diag_bwd<<<1, 64, 0, stream>>>(A, N, kb, B, 64);
    if (kb > 0) rhs_bwd_update<<<kb / 16, 128, 0, stream>>>(A, N, kb, B);
  }
}

static void chol_solve1(float* A, int N, float* bv, hipStream_t stream) {
  for (int kb = 0; kb < N; kb += NB) {
    rhs_diag_fwd<<<1, 64, 0, stream>>>(A, N, kb, bv, 1);
    int M = N - kb - NB;
    if (M > 0) vec_fwd_update<<<(M + 255) / 256, 256, 0, stream>>>(A, N, kb, bv);
  }
  for (int kb = N - NB; kb >= 0; kb -= NB) {
    rhs_diag_bwd<<<1, 64, 0, stream>>>(A, N, kb, bv, 1);
    if (kb > 0) vec_bwd_update<<<(kb + 255) / 256, 256, 0, stream>>>(A, N, kb, bv);
  }
}

extern "C" void kernel_launch(void* const* d_in, const int* in_sizes, int n_in,
                              void* d_out, int out_size, void* d_ws,
                              size_t ws_size, hipStream_t stream) {
  const float* f_x = (const float*)d_in[0];
  const float* x_grid = (const float*)d_in[1];
  const float* y_grid = (const float*)d_in[2];
  const float* qpts = (const float*)d_in[3];
  const float* w = (const float*)d_in[4];
  const float* lift_W = (const float*)d_in[5];
  const float* lift_b = (const float*)d_in[6];
  const float* s_in_raw = (const float*)d_in[7];
  const float* s_out_raw = (const float*)d_in[8];
  const float* int_scales = (const float*)d_in[9];
  const float* conv_W = (const float*)d_in[10];
  const float* conv_b = (const float*)d_in[11];
  const float* p0_W = (const float*)d_in[12];
  const float* p0_b = (const float*)d_in[13];
  const float* p1_W = (const float*)d_in[14];
  const float* p1_b = (const float*)d_in[15];
  const float* p2_W = (const float*)d_in[16];
  const float* p2_b = (const float*)d_in[17];
  float* out = (float*)d_out;

  float* ws = (float*)d_ws;
  float* Kxx = ws;                                  // 4096*4096
  float* Kxq = Kxx + (size_t)N1 * N1;               // 4096*1024
  float* FX = Kxq + (size_t)N1 * NQ;                // 4096*64 (solved in place)
  float* FQA = FX + (size_t)N1 * CH;                // 1024*64
  float* FQB = FQA + (size_t)NQ * CH;               // 1024*64
  float* Rqq = FQB + (size_t)NQ * CH;               // 1024*1024
  float* Kqq = Rqq + (size_t)NQ * NQ;               // 1024*1024
  float* Hv = Kqq + (size_t)NQ * NQ;                // 1024
  float* Kyq = Kxx;  // alias: Kxx dead by stage 3

  // Stage 1: lift, Gram matrices, f_q = k_xq^T (k_xx^{-1} fx)
  lift_kernel<<<N1 * CH / 256, 256, 0, stream>>>(f_x, x_grid, lift_W, lift_b, FX);
  build_kxx<<<dim3(N1 / 16, N1 / 16), dim3(16, 16), 0, stream>>>(Kxx, x_grid,
                                                                 s_in_raw, N1);
  build_kxq<<<dim3(NQ / 16, N1 / 16), dim3(16, 16), 0, stream>>>(Kxq, x_grid,
                                                                 qpts, s_in_raw);
  cholesky_t<N1>(Kxx, stream);
  chol_solve64(Kxx, N1, FX, stream);
  gemm_AtB<<<NQ / 16, 128, 0, stream>>>(Kxq, FX, FQA, N1);
  gelu_kernel<<<NQ * CH / 256, 256, 0, stream>>>(FQA, NQ * CH);

  // Stage 2: integral layers (per-channel Matérn Gram matvec, fused)
  build_rqq<<<dim3(NQ / 16, NQ / 16), dim3(16, 16), 0, stream>>>(Rqq, qpts);
  float* fa = FQA;
  float* fb = FQB;
  for (int l = 0; l < DEPTH; l++) {
    kno_layer<<<NQ, 256, 0, stream>>>(fa, fb, Rqq, conv_W + (size_t)l * CH * CH,
                                      conv_b + (size_t)l * CH,
                                      int_scales + (size_t)l * CH, w,
                                      (l < DEPTH - 1) ? 1 : 0);
    float* tmp = fa; fa = fb; fb = tmp;
  }
  // result back in FQA

  // MLP head
  dense64<<<NQ * CH / 256, 256, 0, stream>>>(fa, p0_W, p0_b, fb, 1);
  dense64<<<NQ * CH / 256, 256, 0, stream>>>(fb, p1_W, p1_b, fa, 1);
  proj1<<<NQ / 256, 256, 0, stream>>>(fa, p2_W, p2_b, Hv);

  // Stage 3: out = k_qy^T (k_qq^{-1} h)
  build_kqq<<<dim3(NQ / 16, NQ / 16), dim3(16, 16), 0, stream>>>(Kqq, Rqq,
                                                                 s_out_raw);
  cholesky_t<NQ>(Kqq, stream);
  chol_solve1(Kqq, NQ, Hv, stream);
  build_kyq<<<dim3(NQ / 16, NY / 16), dim3(16, 16), 0, stream>>>(Kyq, y_grid,
                                                                 qpts, s_out_raw);
  final_out<<<NY, 256, 0, stream>>>(Kyq, Hv, out);
}